// GAT_39393440039564
// MI455X (gfx1250) — compile-verified
//
#include <hip/hip_runtime.h>
#include <math.h>

// ---------------------------------------------------------------------------
// GAT (2-layer, PyG-style with self-loops) for MI455X / gfx1250.
// N=50000 nodes, E=800000 edges (+N self loops), IN=128, HID=32, HEADS=8, OUT=64
// Roofline: edge gather/scatter dominates (~2 GB @ 23.3 TB/s); GEMMs are tiny,
// done in exact-fp32 WMMA (V_WMMA_F32_16X16X4_F32) with async-DMA A-tiles.
// ---------------------------------------------------------------------------

typedef __attribute__((ext_vector_type(2))) float v2f;
typedef __attribute__((ext_vector_type(8))) float v8f;

#define NEG_SLOPE 0.2f

// ---------------------------------------------------------------------------
// FP32 WMMA GEMM:  C[M x NCOL] = A[M x K] * B[K x NCOL]
// One 16-row tile per block. Full tiles stage A via GLOBAL_LOAD_ASYNC_TO_LDS_B128
// (ASYNCcnt-tracked global->LDS DMA, no VGPR round trip). NW waves each own a
// 16x16 C tile and run a K-loop of V_WMMA_F32_16X16X4_F32.
//
// A-fragment (32-bit A 16x4, ISA 7.12.2): lane L holds row M=L&15,
//   vgpr0/1 = K = (L>>4)*2 + {0,1}.
// B-fragment (4x16, rows striped across lanes): lane L holds col N=L&15,
//   vgpr0/1 = K = (L>>4)*2 + {0,1}.
// C/D (16x16 f32): vgpr v: lanes0-15 -> M=v, lanes16-31 -> M=v+8, N=lane&15.
// ---------------------------------------------------------------------------
template <int K, int NW>
__global__ __launch_bounds__(NW * 32)
void wmma_gemm_f32(const float* __restrict__ A, const float* __restrict__ B,
                   float* __restrict__ C, int M, int NCOL) {
  // +4-float row pad: rows stay 16B-aligned for B128 LDS writes, and the
  // 16-lane column reads hit banks (4m+c)%64 -> conflict-free.
  constexpr int PAD = 4;
  __shared__ float lds_a[16 * (K + PAD)];  // sole shared object -> LDS offset 0

  const int row0 = blockIdx.x * 16;
  const int wave = threadIdx.x >> 5;
  const int lane = threadIdx.x & 31;
  const int col0 = (blockIdx.y * NW + wave) * 16;
  const bool full = (row0 + 16 <= M);  // uniform across the block

  if (full) {
    // Async DMA: each thread issues 16B global->LDS copies.
    for (int ci = threadIdx.x; ci < 16 * K / 4; ci += NW * 32) {
      int r = ci / (K / 4);
      int c = (ci % (K / 4)) * 4;
      const float* gp = A + (size_t)(row0 + r) * K + c;
      unsigned lofs = (unsigned)((r * (K + PAD) + c) * 4);
      asm volatile("global_load_async_to_lds_b128 %0, %1, off"
                   :: "v"(lofs), "v"(gp) : "memory");
    }
    asm volatile("s_wait_asynccnt 0x0" ::: "memory");
  } else {
    // Generic edge path (never taken for M % 16 == 0).
    for (int i = threadIdx.x; i < 16 * K; i += NW * 32) {
      int r = i / K, c = i % K;
      float v = 0.0f;
      if (row0 + r < M) v = A[(size_t)(row0 + r) * K + c];
      lds_a[r * (K + PAD) + c] = v;
    }
  }
  __syncthreads();

  const int m  = lane & 15;        // row within A-frag / col within B-frag
  const int kb = (lane >> 4) * 2;  // K sub-offset selected by lane half

  v8f acc = {};
  for (int k = 0; k < K; k += 4) {
    v2f a, b;
    a.x = lds_a[m * (K + PAD) + k + kb];
    a.y = lds_a[m * (K + PAD) + k + kb + 1];
    b.x = B[(size_t)(k + kb) * NCOL + col0 + m];
    b.y = B[(size_t)(k + kb + 1) * NCOL + col0 + m];
    acc = __builtin_amdgcn_wmma_f32_16x16x4_f32(
        /*neg_a=*/false, a, /*neg_b=*/false, b,
        /*c_mod=*/(short)0, acc, /*reuse_a=*/false, /*reuse_b=*/false);
  }

  const int n = col0 + (lane & 15);
  const int rbase = row0 + ((lane >> 4) * 8);
  if (full) {
#pragma unroll
    for (int v = 0; v < 8; ++v)
      C[(size_t)(rbase + v) * NCOL + n] = acc[v];
  } else {
#pragma unroll
    for (int v = 0; v < 8; ++v)
      if (rbase + v < M) C[(size_t)(rbase + v) * NCOL + n] = acc[v];
  }
}

// ---------------------------------------------------------------------------
// Per-(node, head) attention coefficients: as[n,h] = sum_c h[n,h,c]*a_s[h,c]
// ---------------------------------------------------------------------------
template <int H, int C>
__global__ __launch_bounds__(256)
void alpha_k(const float* __restrict__ feat, const float* __restrict__ a_s,
             const float* __restrict__ a_d, float* __restrict__ as,
             float* __restrict__ ad, int N) {
  int t = blockIdx.x * blockDim.x + threadIdx.x;
  if (t >= N * H) return;
  int n = t / H, h = t % H;
  const float* f = feat + (size_t)n * H * C + h * C;
  float s0 = 0.0f, s1 = 0.0f;
#pragma unroll 4
  for (int c = 0; c < C; ++c) {
    float fv = f[c];
    s0 += fv * a_s[h * C + c];
    s1 += fv * a_d[h * C + c];
  }
  as[t] = s0;
  ad[t] = s1;
}

// ---------------------------------------------------------------------------
// Float atomic max via monotonic int/uint bit trick (target init = -inf).
// ---------------------------------------------------------------------------
__device__ inline void atomicMaxF(float* addr, float val) {
  if (val >= 0.0f)
    atomicMax((int*)addr, __float_as_int(val));
  else
    atomicMin((unsigned int*)addr, (unsigned int)__float_as_int(val));
}

__device__ inline void edge_endpoints(const long long* __restrict__ ei, int E,
                                      int e, int& s, int& d) {
  if (e < E) {
    s = (int)ei[e];
    d = (int)ei[(size_t)E + e];
  } else {  // virtual self-loop edge
    s = e - E;
    d = e - E;
  }
}

__device__ inline float leaky(float v) {
  return v > 0.0f ? v : NEG_SLOPE * v;
}

// Pass 1: segment max of leaky_relu(as[src]+ad[dst]) over dst
template <int H>
__global__ __launch_bounds__(256)
void edge_max_k(const long long* __restrict__ ei, int E, int N,
                const float* __restrict__ as, const float* __restrict__ ad,
                float* __restrict__ m) {
  int e = blockIdx.x * blockDim.x + threadIdx.x;
  if (e >= E + N) return;
  int s, d;
  edge_endpoints(ei, E, e, s, d);
#pragma unroll
  for (int h = 0; h < H; ++h) {
    float v = leaky(as[s * H + h] + ad[d * H + h]);
    atomicMaxF(&m[d * H + h], v);
  }
}

// Pass 2: segment sum of exp(e - m[dst]) over dst
template <int H>
__global__ __launch_bounds__(256)
void edge_den_k(const long long* __restrict__ ei, int E, int N,
                const float* __restrict__ as, const float* __restrict__ ad,
                const float* __restrict__ m, float* __restrict__ den) {
  int e = blockIdx.x * blockDim.x + threadIdx.x;
  if (e >= E + N) return;
  int s, d;
  edge_endpoints(ei, E, e, s, d);
#pragma unroll
  for (int h = 0; h < H; ++h) {
    float v = leaky(as[s * H + h] + ad[d * H + h]);
    float ex = expf(v - m[d * H + h]);
    atomicAdd(&den[d * H + h], ex);
  }
}

// Pass 3: agg[dst] += alpha * feat[src]. Wave-per-edge, VEC floats per lane:
// full-width coalesced b128/b64 gathers of feat[src], coalesced f32 atomics.
template <int H, int C, int VEC>
__global__ __launch_bounds__(256)
void edge_agg_k(const long long* __restrict__ ei, int E, int N,
                const float* __restrict__ as, const float* __restrict__ ad,
                const float* __restrict__ m, const float* __restrict__ den,
                const float* __restrict__ feat, float* __restrict__ agg) {
  constexpr int LD = H * C;
  const int wid = (blockIdx.x * blockDim.x + threadIdx.x) >> 5;
  const int lane = threadIdx.x & 31;
  const int nw = (gridDim.x * blockDim.x) >> 5;
  const int ET = E + N;

  for (int e = wid; e < ET; e += nw) {
    int s, d;
    edge_endpoints(ei, E, e, s, d);
#pragma unroll
    for (int it = 0; it < LD / (32 * VEC); ++it) {
      int base = it * 32 * VEC + lane * VEC;       // VEC-chunk within one head
      int h = base / C;
      float v = leaky(as[s * H + h] + ad[d * H + h]);
      float alpha = expf(v - m[d * H + h]) / (den[d * H + h] + 1e-16f);
      const float* fp = feat + (size_t)s * LD + base;
      float* ap = agg + (size_t)d * LD + base;
      if constexpr (VEC == 4) {
        float4 f = *(const float4*)fp;
        atomicAdd(ap + 0, f.x * alpha);
        atomicAdd(ap + 1, f.y * alpha);
        atomicAdd(ap + 2, f.z * alpha);
        atomicAdd(ap + 3, f.w * alpha);
      } else {
        float2 f = *(const float2*)fp;
        atomicAdd(ap + 0, f.x * alpha);
        atomicAdd(ap + 1, f.y * alpha);
      }
    }
  }
}

// ---------------------------------------------------------------------------
// Small utility kernels
// ---------------------------------------------------------------------------
__global__ void fill_k(float* __restrict__ p, float v, size_t n) {
  size_t i = (size_t)blockIdx.x * blockDim.x + threadIdx.x;
  size_t stride = (size_t)gridDim.x * blockDim.x;
  for (; i < n; i += stride) p[i] = v;
}

__global__ void bias_init_k(float* __restrict__ out, const float* __restrict__ b,
                            size_t n, int LD) {
  size_t i = (size_t)blockIdx.x * blockDim.x + threadIdx.x;
  size_t stride = (size_t)gridDim.x * blockDim.x;
  for (; i < n; i += stride) out[i] = b[i % LD];
}

__global__ void elu_k(const float* __restrict__ in, float* __restrict__ out,
                      size_t n) {
  size_t i = (size_t)blockIdx.x * blockDim.x + threadIdx.x;
  size_t stride = (size_t)gridDim.x * blockDim.x;
  for (; i < n; i += stride) {
    float v = in[i];
    out[i] = v > 0.0f ? v : expm1f(v);
  }
}

// ---------------------------------------------------------------------------
// Host launcher
// ---------------------------------------------------------------------------
extern "C" void kernel_launch(void* const* d_in, const int* in_sizes, int n_in,
                              void* d_out, int out_size, void* d_ws, size_t ws_size,
                              hipStream_t stream) {
  constexpr int IN = 128, HID = 32, HEADS = 8, OUT = 64;
  constexpr int LD1 = HEADS * HID;  // 256

  const float*     x   = (const float*)d_in[0];
  const long long* ei  = (const long long*)d_in[1];  // int64 edge_index [2,E]
  const float*     W1  = (const float*)d_in[2];      // [128,256]
  const float*     aS1 = (const float*)d_in[3];      // [8,32]
  const float*     aD1 = (const float*)d_in[4];
  const float*     b1  = (const float*)d_in[5];      // [256]
  const float*     W2  = (const float*)d_in[6];      // [256,64]
  const float*     aS2 = (const float*)d_in[7];      // [1,64]
  const float*     aD2 = (const float*)d_in[8];
  const float*     b2  = (const float*)d_in[9];      // [64]
  float* out = (float*)d_out;                        // [N,64]

  const int N  = in_sizes[0] / IN;
  const int E  = in_sizes[1] / 2;
  const int ET = E + N;  // edges + self loops

  // Workspace carve-up (floats). Total = N*(256+256+64+32+4) ~ 122 MB.
  float* ws = (float*)d_ws;
  size_t off = 0;
  float* h1   = ws + off; off += (size_t)N * LD1;   // features L1 (reused as ELU out)
  float* agg1 = ws + off; off += (size_t)N * LD1;   // aggregation L1 (seeded w/ b1)
  float* h2   = ws + off; off += (size_t)N * OUT;   // features L2
  float* as1  = ws + off; off += (size_t)N * HEADS;
  float* ad1  = ws + off; off += (size_t)N * HEADS;
  float* m1   = ws + off; off += (size_t)N * HEADS;
  float* dn1  = ws + off; off += (size_t)N * HEADS;
  float* as2  = ws + off; off += (size_t)N;
  float* ad2  = ws + off; off += (size_t)N;
  float* m2   = ws + off; off += (size_t)N;
  float* dn2  = ws + off; off += (size_t)N;
  (void)ws_size; (void)n_in; (void)out_size;

  const int mtiles = (N + 15) / 16;

  // ---------------- Layer 1 ----------------
  // h1 = x @ W1   (M=N, K=128, NCOL=256): 8 waves/block cover 128 cols, grid.y=2
  {
    dim3 g(mtiles, LD1 / (8 * 16));
    wmma_gemm_f32<IN, 8><<<g, 256, 0, stream>>>(x, W1, h1, N, LD1);
  }
  alpha_k<HEADS, HID><<<((size_t)N * HEADS + 255) / 256, 256, 0, stream>>>(
      h1, aS1, aD1, as1, ad1, N);

  fill_k<<<1024, 256, 0, stream>>>(m1, -INFINITY, (size_t)N * HEADS);
  fill_k<<<1024, 256, 0, stream>>>(dn1, 0.0f, (size_t)N * HEADS);
  bias_init_k<<<2048, 256, 0, stream>>>(agg1, b1, (size_t)N * LD1, LD1);

  edge_max_k<HEADS><<<(ET + 255) / 256, 256, 0, stream>>>(ei, E, N, as1, ad1, m1);
  edge_den_k<HEADS><<<(ET + 255) / 256, 256, 0, stream>>>(ei, E, N, as1, ad1, m1, dn1);
  edge_agg_k<HEADS, HID, 4><<<(ET + 7) / 8, 256, 0, stream>>>(
      ei, E, N, as1, ad1, m1, dn1, h1, agg1);

  // h1 <- elu(agg1 (+b1 pre-seeded))   (h1 pre-features no longer needed)
  elu_k<<<2048, 256, 0, stream>>>(agg1, h1, (size_t)N * LD1);

  // ---------------- Layer 2 ----------------
  // h2 = h1 @ W2   (M=N, K=256, NCOL=64): 4 waves/block cover all 64 cols
  {
    dim3 g(mtiles, 1);
    wmma_gemm_f32<LD1, 4><<<g, 128, 0, stream>>>(h1, W2, h2, N, OUT);
  }
  alpha_k<1, OUT><<<((size_t)N + 255) / 256, 256, 0, stream>>>(
      h2, aS2, aD2, as2, ad2, N);

  fill_k<<<1024, 256, 0, stream>>>(m2, -INFINITY, (size_t)N);
  fill_k<<<1024, 256, 0, stream>>>(dn2, 0.0f, (size_t)N);
  bias_init_k<<<2048, 256, 0, stream>>>(out, b2, (size_t)N * OUT, OUT);

  edge_max_k<1><<<(ET + 255) / 256, 256, 0, stream>>>(ei, E, N, as2, ad2, m2);
  edge_den_k<1><<<(ET + 255) / 256, 256, 0, stream>>>(ei, E, N, as2, ad2, m2, dn2);
  edge_agg_k<1, OUT, 2><<<(ET + 7) / 8, 256, 0, stream>>>(
      ei, E, N, as2, ad2, m2, dn2, h2, out);
}